// BertCRFModel_20761871909330
// MI455X (gfx1250) — compile-verified
//
#include <hip/hip_runtime.h>
#include <hip/hip_bf16.h>
#include <math.h>

#define Bdim 64
#define Sdim 512
#define Hdim 768
#define Ldim 9

typedef float v2f __attribute__((ext_vector_type(2)));
typedef float v8f __attribute__((ext_vector_type(8)));

// ---------------------------------------------------------------------------
// Kernel 1: emissions = (HS @ W + bias) * class_weights      (B*S x 9)
// WMMA f32 16x16x4.  Each wave owns 64 rows (4 x 16-row tiles), W is staged
// in LDS padded to N=16.  A-fragment layout (fp32 16x4): lanes 0-15 hold
// K=0,1 ; lanes 16-31 hold K=2,3 ; row = lane & 15.  B-fragment: VGPR0 =
// row K (+2 for upper half-wave), VGPR1 = row K+1, col = lane & 15.
// ---------------------------------------------------------------------------
__global__ __launch_bounds__(256) void emis_gemm_wmma(
    const float* __restrict__ hs, const float* __restrict__ W,
    const float* __restrict__ bias, const float* __restrict__ cw,
    float* __restrict__ emis)
{
    __shared__ float ldsW[Hdim * 16];   // 48 KB, zero-padded cols 9..15

    const int tid = threadIdx.x;
    for (int idx = tid; idx < Hdim * 16; idx += 256) {
        const int k = idx >> 4, n = idx & 15;
        ldsW[idx] = (n < Ldim) ? W[k * Ldim + n] : 0.0f;
    }
    __syncthreads();

    const int lane = tid & 31;
    const int wave = tid >> 5;
    const int half = lane >> 4;      // selects K pair for A / B fragments
    const int mr   = lane & 15;      // A row within tile; also N for B/C/D
    const long m0  = (long)blockIdx.x * 512 + (long)wave * 64;

    const float* arow0 = hs + (m0 +  0 + mr) * Hdim;
    const float* arow1 = hs + (m0 + 16 + mr) * Hdim;
    const float* arow2 = hs + (m0 + 32 + mr) * Hdim;
    const float* arow3 = hs + (m0 + 48 + mr) * Hdim;

    v8f acc0 = {}, acc1 = {}, acc2 = {}, acc3 = {};

    for (int k0 = 0; k0 < Hdim; k0 += 4) {
        const int ka = k0 + 2 * half;

        if ((k0 & 31) == 0 && (k0 + 32) < Hdim) {
            __builtin_prefetch(arow0 + k0 + 32, 0, 1);
            __builtin_prefetch(arow1 + k0 + 32, 0, 1);
            __builtin_prefetch(arow2 + k0 + 32, 0, 1);
            __builtin_prefetch(arow3 + k0 + 32, 0, 1);
        }

        const v2f a0 = *(const v2f*)(arow0 + ka);
        const v2f a1 = *(const v2f*)(arow1 + ka);
        const v2f a2 = *(const v2f*)(arow2 + ka);
        const v2f a3 = *(const v2f*)(arow3 + ka);

        v2f bf;
        bf.x = ldsW[(ka + 0) * 16 + mr];
        bf.y = ldsW[(ka + 1) * 16 + mr];

        acc0 = __builtin_amdgcn_wmma_f32_16x16x4_f32(false, a0, false, bf,
                                                     (short)0, acc0, false, false);
        acc1 = __builtin_amdgcn_wmma_f32_16x16x4_f32(false, a1, false, bf,
                                                     (short)0, acc1, false, false);
        acc2 = __builtin_amdgcn_wmma_f32_16x16x4_f32(false, a2, false, bf,
                                                     (short)0, acc2, false, false);
        acc3 = __builtin_amdgcn_wmma_f32_16x16x4_f32(false, a3, false, bf,
                                                     (short)0, acc3, false, false);
    }

    // Epilogue: C/D layout -> lane holds (M = 8*half + r, N = mr), r = VGPR idx.
    if (mr < Ldim) {
        const float bn = bias[mr];
        const float cn = cw[mr];
        #pragma unroll
        for (int r = 0; r < 8; ++r) {
            const long row = m0 + 8 * half + r;
            emis[(row +  0) * Ldim + mr] = (acc0[r] + bn) * cn;
            emis[(row + 16) * Ldim + mr] = (acc1[r] + bn) * cn;
            emis[(row + 32) * Ldim + mr] = (acc2[r] + bn) * cn;
            emis[(row + 48) * Ldim + mr] = (acc3[r] + bn) * cn;
        }
    }
}

// ---------------------------------------------------------------------------
// Kernel 2: CRF NLL pieces per batch element.  One wave32 per sequence.
// Lanes 0..8 carry the 9 forward scores; numerator is time-parallel.
// ---------------------------------------------------------------------------
__global__ __launch_bounds__(32) void crf_forward(
    const float* __restrict__ emis, const float* __restrict__ startT,
    const float* __restrict__ endT, const float* __restrict__ trans,
    const int* __restrict__ labels, const int* __restrict__ attn,
    float* __restrict__ llh)
{
    const int b    = blockIdx.x;
    const int lane = threadIdx.x;
    const int j    = (lane < Ldim) ? lane : 0;

    const float* eb = emis + (size_t)b * Sdim * Ldim;
    const int*   lb = labels + b * Sdim;
    const int*   ab = attn   + b * Sdim;

    // transitions column j cached in registers
    float tc[Ldim];
    #pragma unroll
    for (int i = 0; i < Ldim; ++i) tc[i] = trans[i * Ldim + j];

    // ---- numerator (parallel over time, strided by lane) ----
    float part = 0.0f;
    int   cnt  = 0;
    for (int t = lane; t < Sdim; t += 32) {
        const int  lab = lb[t];
        const int  tag = (lab == -100) ? 0 : lab;
        const bool mk  = (t == 0) || ((lab != -100) && (ab[t] == 1));
        cnt += mk ? 1 : 0;
        if (t == 0) {
            part += startT[tag] + eb[tag];
        } else if (mk) {
            const int labp = lb[t - 1];
            const int tagp = (labp == -100) ? 0 : labp;
            part += trans[tagp * Ldim + tag] + eb[t * Ldim + tag];
        }
    }
    #pragma unroll
    for (int off = 16; off >= 1; off >>= 1) {
        part += __shfl_xor(part, off, 32);
        cnt  += __shfl_xor(cnt,  off, 32);
    }
    const int  last = cnt - 1;
    const int  labl = lb[last];
    const int  tagl = (labl == -100) ? 0 : labl;
    const float numer = part + endT[tagl];

    // ---- forward recursion (sequential in t, parallel over states) ----
    float score = startT[j] + eb[j];
    for (int t = 1; t < Sdim; ++t) {
        const int  lab = lb[t];
        const bool mk  = (lab != -100) && (ab[t] == 1);
        const float e  = eb[t * Ldim + j];

        float v[Ldim];
        float m = -3.402823e38f;
        #pragma unroll
        for (int i = 0; i < Ldim; ++i) {
            const float si = __shfl(score, i, 32);
            v[i] = si + tc[i];
            m = fmaxf(m, v[i]);
        }
        float s = 0.0f;
        #pragma unroll
        for (int i = 0; i < Ldim; ++i) s += __expf(v[i] - m);
        const float nxt = m + __logf(s) + e;
        score = mk ? nxt : score;
    }

    // log_z = logsumexp over lanes 0..8 of (score + end)
    const float fin = score + endT[j];
    const float fv  = (lane < Ldim) ? fin : -3.402823e38f;
    float m2 = fv;
    #pragma unroll
    for (int off = 16; off >= 1; off >>= 1) m2 = fmaxf(m2, __shfl_xor(m2, off, 32));
    float s2 = __expf(fv - m2);
    #pragma unroll
    for (int off = 16; off >= 1; off >>= 1) s2 += __shfl_xor(s2, off, 32);
    const float logz = m2 + __logf(s2);

    if (lane == 0) llh[b] = numer - logz;
}

// ---------------------------------------------------------------------------
// Kernel 3: out = -mean(llh)
// ---------------------------------------------------------------------------
__global__ __launch_bounds__(64) void reduce_mean_neg(
    const float* __restrict__ llh, float* __restrict__ out)
{
    __shared__ float sh[64];
    const int t = threadIdx.x;
    sh[t] = llh[t];
    __syncthreads();
    #pragma unroll
    for (int off = 32; off >= 1; off >>= 1) {
        if (t < off) sh[t] += sh[t + off];
        __syncthreads();
    }
    if (t == 0) out[0] = -sh[0] / (float)Bdim;
}

extern "C" void kernel_launch(void* const* d_in, const int* in_sizes, int n_in,
                              void* d_out, int out_size, void* d_ws, size_t ws_size,
                              hipStream_t stream) {
    const float* hs     = (const float*)d_in[0];
    const float* W      = (const float*)d_in[1];
    const float* bias   = (const float*)d_in[2];
    const float* cwght  = (const float*)d_in[3];
    const float* startT = (const float*)d_in[4];
    const float* endT   = (const float*)d_in[5];
    const float* trans  = (const float*)d_in[6];
    const int*   labels = (const int*)d_in[7];
    const int*   attn   = (const int*)d_in[8];

    float* emis = (float*)d_ws;                         // B*S*9 floats
    float* llh  = emis + (size_t)Bdim * Sdim * Ldim;    // B floats

    emis_gemm_wmma<<<64, 256, 0, stream>>>(hs, W, bias, cwght, emis);
    crf_forward<<<Bdim, 32, 0, stream>>>(emis, startT, endT, trans, labels, attn, llh);
    reduce_mean_neg<<<1, 64, 0, stream>>>(llh, (float*)d_out);
    (void)in_sizes; (void)n_in; (void)out_size; (void)ws_size;
}